// KMeans_3161095930011
// MI455X (gfx1250) — compile-verified
//
#include <hip/hip_runtime.h>

typedef float v2f __attribute__((ext_vector_type(2)));
typedef float v8f __attribute__((ext_vector_type(8)));

#define HW        65536     // 256*256
#define NCH       3
#define NCLUSTER  512
#define NTILES    (NCLUSTER / 16)      // 32 cluster tiles
#define WAVES_PB  8
#define BLOCK     (WAVES_PB * 32)      // 256 threads = 8 wave32
#define PIXTILES  2                    // pixel tiles (of 16) per wave

// Each wave handles 32 pixels (two 16-row A tiles) against all 512 clusters.
// Score via v_wmma_f32_16x16x4_f32:  D[m][n] = sum_k A[m][k]*B[k][n]
//   A row m = [x0, x1, x2, 1]
//   B col n = [-2*C_n0, -2*C_n1, -2*C_n2, ||C_n||^2]
// => D[m][n] = -2*x_m.C_n + ||C_n||^2  (argmin-equivalent to squared dist;
// the per-pixel ||x||^2 term is constant and dropped).
__global__ __launch_bounds__(BLOCK) void kmeans_assign_kernel(
    const float* __restrict__ x,   // [4, 3, HW]
    const float* __restrict__ C,   // [NCLUSTER, 3]
    float* __restrict__ out,       // [4*HW] indices as float
    int npix)
{
    // ---- build B tiles in LDS (shared by all 8 waves of the block) ----
    // B-matrix f32 4x16 layout (wave32): lane n<16 -> {K0,K1} of col n,
    // lane n+16 -> {K2,K3} of col n.
    __shared__ v2f Bs[NTILES * 32];   // 8 KB
    for (int i = threadIdx.x; i < NTILES * 32; i += BLOCK) {
        int ln = i & 31;
        int k  = (i >> 5) * 16 + (ln & 15);
        float c0 = C[k * NCH + 0];
        float c1 = C[k * NCH + 1];
        float c2 = C[k * NCH + 2];
        v2f b;
        if (ln < 16) { b.x = -2.0f * c0; b.y = -2.0f * c1; }
        else         { b.x = -2.0f * c2; b.y = c0*c0 + c1*c1 + c2*c2; }
        Bs[i] = b;
    }
    __syncthreads();

    const int lane = threadIdx.x & 31;
    const int wave = threadIdx.x >> 5;
    const int pixbase = (blockIdx.x * WAVES_PB + wave) * (16 * PIXTILES);
    if (pixbase >= npix) return;

    // ---- load A tiles (f32 16x4): lanes 0-15 carry K0/K1 of row M=lane,
    // lanes 16-31 carry K2/K3 of row M=lane-16.
    // 32 | pixbase and 32 | HW so both tiles sit in one batch image.
    const int bb = pixbase >> 16;              // batch index (HW = 65536)
    const int pp = pixbase & (HW - 1);
    const float* xb = x + (size_t)bb * (NCH * HW) + pp + (lane & 15);
    v2f a0, a1;
    if (lane < 16) {
        a0.x = xb[0];           a0.y = xb[HW];        // ch0, ch1 (pixels 0..15)
        a1.x = xb[16];          a1.y = xb[HW + 16];   // ch0, ch1 (pixels 16..31)
    } else {
        a0.x = xb[2 * HW];      a0.y = 1.0f;          // ch2, bias
        a1.x = xb[2 * HW + 16]; a1.y = 1.0f;
    }

    // running (min value, argmin cluster) per D row held by this lane
    float bv0[8], bv1[8];
    int   bi0[8], bi1[8];
#pragma unroll
    for (int j = 0; j < 8; ++j) {
        bv0[j] = __builtin_inff(); bi0[j] = 0;
        bv1[j] = __builtin_inff(); bi1[j] = 0;
    }

#pragma unroll 2
    for (int t = 0; t < NTILES; ++t) {
        v2f bmat = Bs[t * 32 + lane];     // one LDS read feeds two WMMAs
        v8f z = {0.f, 0.f, 0.f, 0.f, 0.f, 0.f, 0.f, 0.f};
        v8f acc0 = __builtin_amdgcn_wmma_f32_16x16x4_f32(
            false, a0, false, bmat, (short)0, z, false, false);
        v8f acc1 = __builtin_amdgcn_wmma_f32_16x16x4_f32(
            false, a1, false, bmat, (short)0, z, false, false);
        int cidx = t * 16 + (lane & 15);  // cluster column this lane holds
#pragma unroll
        for (int j = 0; j < 8; ++j) {
            float v0 = acc0[j];
            if (v0 < bv0[j]) { bv0[j] = v0; bi0[j] = cidx; }  // strict <: first-min wins
            float v1 = acc1[j];
            if (v1 < bv1[j]) { bv1[j] = v1; bi1[j] = cidx; }
        }
    }

    // ---- cross-lane argmin across the 16 columns of each half-wave ----
    // D layout: lanes 0-15 hold rows M=0..7, lanes 16-31 hold rows M=8..15;
    // xor masks 1,2,4,8 stay inside each 16-lane half.
#pragma unroll
    for (int off = 1; off < 16; off <<= 1) {
#pragma unroll
        for (int j = 0; j < 8; ++j) {
            float ov0 = __shfl_xor(bv0[j], off, 32);
            int   oi0 = __shfl_xor(bi0[j], off, 32);
            if (ov0 < bv0[j] || (ov0 == bv0[j] && oi0 < bi0[j])) { bv0[j] = ov0; bi0[j] = oi0; }
            float ov1 = __shfl_xor(bv1[j], off, 32);
            int   oi1 = __shfl_xor(bi1[j], off, 32);
            if (ov1 < bv1[j] || (ov1 == bv1[j] && oi1 < bi1[j])) { bv1[j] = ov1; bi1[j] = oi1; }
        }
    }

    // lane 0 writes rows m=0..7, lane 16 writes rows m=8..15 (per pixel tile)
    if ((lane & 15) == 0) {
        int mbase = (lane >> 4) * 8;
#pragma unroll
        for (int j = 0; j < 8; ++j) {
            out[pixbase + mbase + j]      = (float)bi0[j];
            out[pixbase + 16 + mbase + j] = (float)bi1[j];
        }
    }
}

extern "C" void kernel_launch(void* const* d_in, const int* in_sizes, int n_in,
                              void* d_out, int out_size, void* d_ws, size_t ws_size,
                              hipStream_t stream) {
    const float* x = (const float*)d_in[0];   // [4,3,256,256] f32
    const float* C = (const float*)d_in[1];   // [512,3] f32
    float* out = (float*)d_out;               // [4*65536] indices as float

    int npix   = in_sizes[0] / NCH;                       // 262144
    int pwaves = npix / (16 * PIXTILES);                  // 8192 waves
    int blocks = (pwaves + WAVES_PB - 1) / WAVES_PB;      // 1024 blocks

    kmeans_assign_kernel<<<blocks, BLOCK, 0, stream>>>(x, C, out, npix);
}